// TensorDot_7232724927058
// MI455X (gfx1250) — compile-verified
//
#include <hip/hip_runtime.h>
#include <cstdint>

typedef __attribute__((ext_vector_type(2))) float v2f;
typedef __attribute__((ext_vector_type(8))) float v8f;

// Output: (N=2048, I=32, C=64, C=64) fp32, 1 GiB -> HBM-write-bound (~50us
// floor at 23.3 TB/s). One block per (n, i). 256 threads = 8 waves; each wave
// computes two 16x16 tiles of the 64x64 output via V_WMMA_F32_16X16X4_F32.
// Staging uses GLOBAL_LOAD_ASYNC_TO_LDS_B128 (ASYNCcnt); output uses
// non-temporal stores (write-once data, keep L2 clean).

__global__ __launch_bounds__(256)
void tensordot_uv_wmma(const float* __restrict__ in1,
                       const float* __restrict__ in2,
                       float* __restrict__ out) {
    // LDS staging: K padded to 8 rows x 64 channels per operand (4 KiB total).
    __shared__ __align__(16) float lds1[8 * 64];
    __shared__ __align__(16) float lds2[8 * 64];

    const int tid = threadIdx.x;
    const int blk = blockIdx.x;       // 0 .. 2048*32-1
    const int i   = blk & 31;         // irrep index: i = l*8 + j
    const int n   = blk >> 5;
    const int l   = i >> 3;
    const int j   = i & 7;
    const int d   = 2 * l + 1;        // 1,3,5,7
    const int off = 8 * l * l;        // row offset of this l-group in DIM=128
    const int row0 = off + j * d;     // first m-row for this (l, j)

    const float scale = (l == 0) ? 1.0f
                      : (l == 1) ? 0.57735026918962576f   // 1/sqrt(3)
                      : (l == 2) ? 0.44721359549995794f   // 1/sqrt(5)
                                 : 0.37796447300922720f;  // 1/sqrt(7)

    // Contiguous global chunk for this (n, i): d*64 floats each operand.
    const float* __restrict__ g1 = in1 + ((size_t)n * 128 + row0) * 64;
    const float* __restrict__ g2 = in2 + ((size_t)n * 128 + row0) * 64;

    // ---- Stage inputs into LDS ----------------------------------------
    // 128 float4 slots per operand. tid 0..127 -> operand 1, 128..255 -> 2
    // (whole-wave uniform). Valid region (slot < d*16): async B128 copy
    // global -> LDS (ASYNCcnt). Pad region: zero-fill via ds stores.
    {
        const int  slot  = tid & 127;
        const bool which = (tid >= 128);
        float*       lds = which ? lds2 : lds1;
        const float* g   = which ? g2   : g1;

        if (slot < d * 16) {
            // LDS aperture maps addr[31:0] directly to the LDS byte offset.
            unsigned lds_off = (unsigned)(uintptr_t)(lds + slot * 4);
            unsigned long long gaddr =
                (unsigned long long)(uintptr_t)(g + slot * 4);
            asm volatile("global_load_async_to_lds_b128 %0, %1, off"
                         :: "v"(lds_off), "v"(gaddr)
                         : "memory");
        } else {
            float4 z = make_float4(0.f, 0.f, 0.f, 0.f);
            *(float4*)(lds + slot * 4) = z;
        }
    }
    asm volatile("s_wait_asynccnt 0x0" ::: "memory");
    __syncthreads();

    // ---- WMMA compute --------------------------------------------------
    const int wave = tid >> 5;        // wave32
    const int lane = tid & 31;
    const int M    = lane & 15;       // row within 16x16 tile (A) / col (B)
    const int half = lane >> 4;       // 0: K={0,1}, 1: K={2,3} per ISA layout
    const int kb   = half * 2;

    // Number of K=4 WMMA steps (uniform per block -> EXEC stays all-ones).
    const int ksteps = (d > 4) ? 2 : 1;

    const size_t outNI = (size_t)(n * 32 + i) * 64 * 64;

    #pragma unroll
    for (int t = 2 * wave; t < 2 * wave + 2; ++t) {
        const int tu = t >> 2;        // tile row (u)
        const int tv = t & 3;         // tile col (v)
        v8f acc = {};

        for (int ks = 0; ks < ksteps; ++ks) {
            const int k0 = ks * 4 + kb;
            v2f a, b;
            // A[M][k] = x1[m=k][u = tu*16 + M]
            a.x = lds1[(k0 + 0) * 64 + tu * 16 + M];
            a.y = lds1[(k0 + 1) * 64 + tu * 16 + M];
            // B[k][N] = x2[m=k][v = tv*16 + N]
            b.x = lds2[(k0 + 0) * 64 + tv * 16 + M];
            b.y = lds2[(k0 + 1) * 64 + tv * 16 + M];
            acc = __builtin_amdgcn_wmma_f32_16x16x4_f32(
                /*neg_a=*/false, a, /*neg_b=*/false, b,
                /*c_mod=*/(short)0, acc,
                /*reuse_a=*/false, /*reuse_b=*/false);
        }

        // D layout: VGPR r -> row u = r + 8*half, col v = tv*16 + (lane%16).
        // Output is write-once -> non-temporal stores (TH=NT), keep L2 clean.
        const size_t colBase = outNI + (size_t)(tv * 16 + M);
        #pragma unroll
        for (int r = 0; r < 8; ++r) {
            const int u = r + 8 * half;
            __builtin_nontemporal_store(acc[r] * scale,
                                        &out[colBase + (size_t)u * 64]);
        }
    }
}

extern "C" void kernel_launch(void* const* d_in, const int* in_sizes, int n_in,
                              void* d_out, int out_size, void* d_ws, size_t ws_size,
                              hipStream_t stream) {
    (void)in_sizes; (void)n_in; (void)out_size; (void)d_ws; (void)ws_size;
    const float* in1 = (const float*)d_in[0];
    const float* in2 = (const float*)d_in[1];
    float* out = (float*)d_out;

    const int N = 2048;               // batch
    const int I = 32;                 // irrep count
    dim3 grid(N * I);                 // one block per (n, i)
    dim3 block(256);                  // 8 waves
    tensordot_uv_wmma<<<grid, block, 0, stream>>>(in1, in2, out);
}